// CRF_29265907155259
// MI455X (gfx1250) — compile-verified
//
#include <hip/hip_runtime.h>
#include <hip/hip_bf16.h>

typedef __attribute__((ext_vector_type(2))) float v2f;
typedef __attribute__((ext_vector_type(8))) float v8f;

#define K_TAGS 48
#define STOP_TAG 47
#define NEG_VAL -10000.0f

// One wave32 per 16 batches. Score state kept in WMMA C/D layout:
//   S[c][r] : tile c (tags 16c..16c+15), VGPR r; batch = r + 8*(lane>=16), tag = 16c + (lane&15)
// Et = exp(trans) resident in registers in B-layout: Bt[s][c], K-slice s (j = 4s..4s+3), N-tile c.
__global__ __launch_bounds__(32) void crf_fwd_kernel(
    const float* __restrict__ h_tag, const float* __restrict__ mask,
    const float* __restrict__ trans, float* __restrict__ out, int T)
{
  const int lane = threadIdx.x & 31;
  const int lo   = lane & 15;
  const int hi   = lane >> 4;
  const int b0   = blockIdx.x * 16;

  __shared__ float p_lds[16 * 52];   // 16 batches x 48 probs, stride 52 -> conflict-free banks

  // ---- Et (constant) into B-layout register tiles: B[k,n] = exp(trans[16c+n][4s+k]) ----
  v2f Bt[12][3];
#pragma unroll
  for (int s = 0; s < 12; ++s)
#pragma unroll
    for (int c = 0; c < 3; ++c) {
      const int i = 16 * c + lo;           // N index (new tag)
      const int j = 4 * s + 2 * hi;        // K index (old tag), lanes 16-31 hold K+2
      v2f b;
      b[0] = __expf(trans[i * K_TAGS + j + 0]);
      b[1] = __expf(trans[i * K_TAGS + j + 1]);
      Bt[s][c] = b;
    }

  // trans[STOP] row for the final add (uniform across r)
  float tS[3];
#pragma unroll
  for (int c = 0; c < 3; ++c) tS[c] = trans[STOP_TAG * K_TAGS + 16 * c + lo];

  // ---- init score0: NEG everywhere, 0 at STOP ----
  float S[3][8];
#pragma unroll
  for (int c = 0; c < 3; ++c)
#pragma unroll
    for (int r = 0; r < 8; ++r)
      S[c][r] = (c == 2 && lo == 15) ? 0.0f : NEG_VAL;

  const size_t strideB = (size_t)T * K_TAGS;

#pragma unroll 1
  for (int t = 0; t < T; ++t) {
    // --- issue this step's emit/mask loads early; a full step of compute hides them ---
    float emit[3][8];
#pragma unroll
    for (int c = 0; c < 3; ++c)
#pragma unroll
      for (int r = 0; r < 8; ++r)
        emit[c][r] = h_tag[(size_t)(b0 + r + 8 * hi) * strideB +
                           (size_t)t * K_TAGS + 16 * c + lo];
    float mrow = mask[(size_t)(b0 + lo) * T + t];   // lane lo holds mask of batch b0+lo
    if (t + 4 < T)  // software prefetch 4 steps ahead (2 points cover the 192B row)
      __builtin_prefetch(&h_tag[(size_t)(b0 + lo) * strideB +
                                (size_t)(t + 4) * K_TAGS + hi * 24], 0, 1);

    // --- per-batch max over 48 tags (per-lane over tiles, then 16-lane xor reduce) ---
    float mr[8];
#pragma unroll
    for (int r = 0; r < 8; ++r) {
      float m = fmaxf(S[0][r], fmaxf(S[1][r], S[2][r]));
#pragma unroll
      for (int d = 1; d <= 8; d <<= 1)
        m = fmaxf(m, __shfl_xor(m, d, 32));
      mr[r] = m;
    }

    // --- p = exp(S - m), transpose D-layout -> A-layout via LDS ---
#pragma unroll
    for (int c = 0; c < 3; ++c)
#pragma unroll
      for (int r = 0; r < 8; ++r)
        p_lds[(r + 8 * hi) * 52 + 16 * c + lo] = __expf(S[c][r] - mr[r]);
    __syncthreads();

    v2f A[12];
#pragma unroll
    for (int s = 0; s < 12; ++s) {      // A: M = lo (batch), K = 4s + v + 2*hi
      v2f a;
      a[0] = p_lds[lo * 52 + 4 * s + 2 * hi + 0];
      a[1] = p_lds[lo * 52 + 4 * s + 2 * hi + 1];
      A[s] = a;
    }

    // --- acc(16x48) = P(16x48) x Et^T(48x48) via 36 chained f32 WMMAs ---
    v8f acc[3];
#pragma unroll
    for (int c = 0; c < 3; ++c) {
      v8f z = {0.f, 0.f, 0.f, 0.f, 0.f, 0.f, 0.f, 0.f};
#pragma unroll
      for (int s = 0; s < 12; ++s)
        z = __builtin_amdgcn_wmma_f32_16x16x4_f32(
                false, A[s], false, Bt[s][c], (short)0, z, false, false);
      acc[c] = z;
    }

    // --- S = mask ? emit + m + log(acc) : S  (select, not blend: avoids 0*inf NaN) ---
#pragma unroll
    for (int r = 0; r < 8; ++r) {
      float mv = __shfl(mrow, r + 8 * hi, 32);   // mask for batch r+8*hi
#pragma unroll
      for (int c = 0; c < 3; ++c) {
        float ns = emit[c][r] + mr[r] + __logf(acc[c][r]);
        S[c][r] = (mv != 0.0f) ? ns : S[c][r];
      }
    }
    __syncthreads();   // protect p_lds WAR before next iteration
  }

  // ---- final: score += trans[STOP]; out[b] = logsumexp_k score[b,k] ----
#pragma unroll
  for (int c = 0; c < 3; ++c)
#pragma unroll
    for (int r = 0; r < 8; ++r) S[c][r] += tS[c];

#pragma unroll
  for (int r = 0; r < 8; ++r) {
    float m = fmaxf(S[0][r], fmaxf(S[1][r], S[2][r]));
#pragma unroll
    for (int d = 1; d <= 8; d <<= 1) m = fmaxf(m, __shfl_xor(m, d, 32));
    float s = __expf(S[0][r] - m) + __expf(S[1][r] - m) + __expf(S[2][r] - m);
#pragma unroll
    for (int d = 1; d <= 8; d <<= 1) s += __shfl_xor(s, d, 32);
    float res = m + __logf(s);
    if (lo == r) out[b0 + r + 8 * hi] = res;   // lanes r and r+16 -> batches r and r+8
  }
}

extern "C" void kernel_launch(void* const* d_in, const int* in_sizes, int n_in,
                              void* d_out, int out_size, void* d_ws, size_t ws_size,
                              hipStream_t stream) {
  const float* h_tag = (const float*)d_in[0];
  const float* mask  = (const float*)d_in[1];
  const float* trans = (const float*)d_in[2];
  float* out = (float*)d_out;

  const int B = out_size;                 // 512
  const int T = in_sizes[1] / B;          // mask is B*T -> 1024

  crf_fwd_kernel<<<dim3(B / 16), dim3(32), 0, stream>>>(h_tag, mask, trans, out, T);
}